// GCN_34565896798643
// MI455X (gfx1250) — compile-verified
//
#include <hip/hip_runtime.h>
#include <math.h>

typedef __attribute__((ext_vector_type(2))) float v2f;
typedef __attribute__((ext_vector_type(8))) float v8f;

// ---------------------------------------------------------------------------
// Degree / normalization
// ---------------------------------------------------------------------------
__global__ void init_deg_kernel(float* deg, int n) {
    int i = blockIdx.x * blockDim.x + threadIdx.x;
    if (i < n) deg[i] = 1.0f;  // self-loop contributes 1 to every node's degree
}

__global__ void acc_deg_kernel(const int* __restrict__ dst, float* deg, int E) {
    int e = blockIdx.x * blockDim.x + threadIdx.x;
    if (e < E) atomicAdd(&deg[dst[e]], 1.0f);
}

__global__ void to_dinv_kernel(float* deg, int n) {
    int i = blockIdx.x * blockDim.x + threadIdx.x;
    if (i < n) deg[i] = rsqrtf(deg[i]);  // deg >= 1 always (self-loops)
}

// ---------------------------------------------------------------------------
// GEMM1: h1[n,16] = x[n,128] @ w1[128,16] using V_WMMA_F32_16X16X4_F32.
// One wave computes one 16-row x 16-col tile; K loop of 128 in steps of 4.
//
// A frag (16x4 f32, ISA 7.12.2): lanes 0-15 hold M=lane, {K0,K1} in v[0],v[1];
//                                lanes 16-31 hold M=lane-16, {K2,K3}.
// B frag (4x16 f32, symmetric):  lanes 0-15 hold N=lane, rows {K0,K1};
//                                lanes 16-31 hold N=lane-16, rows {K2,K3}.
// C/D (16x16 f32): VGPR r, lanes 0-15 -> (M=r, N=lane);
//                  lanes 16-31 -> (M=8+r, N=lane-16).
// ---------------------------------------------------------------------------
__global__ void gemm1_wmma_kernel(const float* __restrict__ x,
                                  const float* __restrict__ w1,
                                  float* __restrict__ h1, int n) {
    int wave = (int)((blockIdx.x * blockDim.x + threadIdx.x) >> 5);
    int lane = threadIdx.x & 31;
    int row0 = wave * 16;
    if (row0 >= n) return;  // whole wave exits together; EXEC stays all-ones

    int m    = lane & 15;        // row (A) / col (B) handled by this lane
    int koff = (lane >> 4) * 2;  // 0 for lanes 0-15, 2 for lanes 16-31

    const float* xrow = x + (size_t)(row0 + m) * 128;

    v8f acc = {};
    for (int k0 = 0; k0 < 128; k0 += 4) {
        v2f a, b;
        a.x = xrow[k0 + koff + 0];
        a.y = xrow[k0 + koff + 1];
        b.x = w1[(k0 + koff + 0) * 16 + m];
        b.y = w1[(k0 + koff + 1) * 16 + m];
        acc = __builtin_amdgcn_wmma_f32_16x16x4_f32(
            /*neg_a=*/false, a, /*neg_b=*/false, b,
            /*c_mod=*/(short)0, acc, /*reuse_a=*/false, /*reuse_b=*/false);
    }

    int nCol  = lane & 15;
    int mBase = (lane >> 4) * 8;
#pragma unroll
    for (int r = 0; r < 8; ++r) {
        h1[(size_t)(row0 + mBase + r) * 16 + nCol] = acc[r];
    }
}

// ---------------------------------------------------------------------------
// Scatter layer 1: out1 = b1 + self_loop + sum_e h1[src]*dinv[src]*dinv[dst]
// ---------------------------------------------------------------------------
__global__ void init_out1_kernel(const float* __restrict__ h1,
                                 const float* __restrict__ dinv,
                                 const float* __restrict__ b1,
                                 float* __restrict__ out1, int n) {
    int t = blockIdx.x * blockDim.x + threadIdx.x;  // (node, feat)
    if (t >= n * 16) return;
    int i = t >> 4, c = t & 15;
    float di = dinv[i];
    out1[t] = b1[c] + h1[t] * di * di;  // self-loop norm = dinv[i]^2
}

__global__ void scatter1_kernel(const int* __restrict__ src,
                                const int* __restrict__ dst,
                                const float* __restrict__ h1,
                                const float* __restrict__ dinv,
                                float* __restrict__ out1, int E) {
    long long t = (long long)blockIdx.x * blockDim.x + threadIdx.x;
    if (t >= (long long)E * 16) return;
    int e = (int)(t >> 4), c = (int)(t & 15);
    int s = src[e], d = dst[e];
    float norm = dinv[s] * dinv[d];
    atomicAdd(&out1[(size_t)d * 16 + c], h1[(size_t)s * 16 + c] * norm);
}

// ---------------------------------------------------------------------------
// GEMM2 fused with ReLU: h2[n,4] = relu(out1)[n,16] @ w2[16,4]
// ---------------------------------------------------------------------------
__global__ void gemm2_relu_kernel(const float* __restrict__ out1,
                                  const float* __restrict__ w2,
                                  float* __restrict__ h2, int n) {
    int i = blockIdx.x * blockDim.x + threadIdx.x;
    if (i >= n) return;
    float v[16];
#pragma unroll
    for (int k = 0; k < 16; ++k) {
        float t = out1[(size_t)i * 16 + k];
        v[k] = t > 0.0f ? t : 0.0f;
    }
#pragma unroll
    for (int c = 0; c < 4; ++c) {
        float s = 0.0f;
#pragma unroll
        for (int k = 0; k < 16; ++k) s += v[k] * w2[k * 4 + c];
        h2[(size_t)i * 4 + c] = s;
    }
}

// ---------------------------------------------------------------------------
// Scatter layer 2
// ---------------------------------------------------------------------------
__global__ void init_out2_kernel(const float* __restrict__ h2,
                                 const float* __restrict__ dinv,
                                 const float* __restrict__ b2,
                                 float* __restrict__ out2, int n) {
    int t = blockIdx.x * blockDim.x + threadIdx.x;  // (node, feat)
    if (t >= n * 4) return;
    int i = t >> 2, c = t & 3;
    float di = dinv[i];
    out2[t] = b2[c] + h2[t] * di * di;
}

__global__ void scatter2_kernel(const int* __restrict__ src,
                                const int* __restrict__ dst,
                                const float* __restrict__ h2,
                                const float* __restrict__ dinv,
                                float* __restrict__ out2, int E) {
    long long t = (long long)blockIdx.x * blockDim.x + threadIdx.x;
    if (t >= (long long)E * 4) return;
    int e = (int)(t >> 2), c = (int)(t & 3);
    int s = src[e], d = dst[e];
    float norm = dinv[s] * dinv[d];
    atomicAdd(&out2[(size_t)d * 4 + c], h2[(size_t)s * 4 + c] * norm);
}

// ---------------------------------------------------------------------------
// log_softmax over 4 classes
// ---------------------------------------------------------------------------
__global__ void log_softmax_kernel(const float* __restrict__ acc,
                                   float* __restrict__ out, int n) {
    int i = blockIdx.x * blockDim.x + threadIdx.x;
    if (i >= n) return;
    float4 a = ((const float4*)acc)[i];
    float m = fmaxf(fmaxf(a.x, a.y), fmaxf(a.z, a.w));
    float s = __expf(a.x - m) + __expf(a.y - m) + __expf(a.z - m) + __expf(a.w - m);
    float lse = m + __logf(s);
    float4 r;
    r.x = a.x - lse; r.y = a.y - lse; r.z = a.z - lse; r.w = a.w - lse;
    ((float4*)out)[i] = r;
}

// ---------------------------------------------------------------------------
extern "C" void kernel_launch(void* const* d_in, const int* in_sizes, int n_in,
                              void* d_out, int out_size, void* d_ws, size_t ws_size,
                              hipStream_t stream) {
    const float* x  = (const float*)d_in[0];
    const int*   ei = (const int*)d_in[1];
    const float* w1 = (const float*)d_in[2];
    const float* b1 = (const float*)d_in[3];
    const float* w2 = (const float*)d_in[4];
    const float* b2 = (const float*)d_in[5];
    float* out = (float*)d_out;

    const int n = in_sizes[0] / 128;  // 100000
    const int E = in_sizes[1] / 2;    // 3200000
    const int* src = ei;
    const int* dst = ei + E;

    // Workspace layout (floats): dinv[n], h1[n*16], out1[n*16], h2[n*4], acc2[n*4]
    float* dinv = (float*)d_ws;
    float* h1   = dinv + n;
    float* o1   = h1 + (size_t)n * 16;
    float* h2   = o1 + (size_t)n * 16;
    float* a2   = h2 + (size_t)n * 4;

    const int B = 256;
    auto cdiv = [](long long a, long long b) { return (int)((a + b - 1) / b); };

    // 1) degrees -> dinv
    init_deg_kernel<<<cdiv(n, B), B, 0, stream>>>(dinv, n);
    acc_deg_kernel<<<cdiv(E, B), B, 0, stream>>>(dst, dinv, E);
    to_dinv_kernel<<<cdiv(n, B), B, 0, stream>>>(dinv, n);

    // 2) h1 = x @ w1  (WMMA f32 16x16x4; n divisible by 16 -> full waves)
    {
        long long threads = (long long)((n + 15) / 16) * 32;
        gemm1_wmma_kernel<<<cdiv(threads, B), B, 0, stream>>>(x, w1, h1, n);
    }

    // 3) layer-1 normalize + scatter-add + bias
    init_out1_kernel<<<cdiv((long long)n * 16, B), B, 0, stream>>>(h1, dinv, b1, o1, n);
    scatter1_kernel<<<cdiv((long long)E * 16, B), B, 0, stream>>>(src, dst, h1, dinv, o1, E);

    // 4) h2 = relu(out1) @ w2
    gemm2_relu_kernel<<<cdiv(n, B), B, 0, stream>>>(o1, w2, h2, n);

    // 5) layer-2 normalize + scatter-add + bias
    init_out2_kernel<<<cdiv((long long)n * 4, B), B, 0, stream>>>(h2, dinv, b2, a2, n);
    scatter2_kernel<<<cdiv((long long)E * 4, B), B, 0, stream>>>(src, dst, h2, dinv, a2, E);

    // 6) log_softmax -> d_out
    log_softmax_kernel<<<cdiv(n, B), B, 0, stream>>>(a2, out, n);
}